// Get_Top_K_10453950398707
// MI455X (gfx1250) — compile-verified
//
#include <hip/hip_runtime.h>
#include <stdint.h>

typedef unsigned int uint;

#define COLS   16384
#define BLOCK  1024
#define EPT    16          // elements per thread = COLS / BLOCK
#define NWAVES 32          // wave32: 1024 / 32

// ---- CDNA5 async global->LDS path (guarded; falls back to direct loads) ----
#if defined(__has_builtin)
#  if __has_builtin(__builtin_amdgcn_global_load_async_to_lds_b128)
#    define HAVE_ASYNC_LDS 1
#  endif
#  if __has_builtin(__builtin_amdgcn_s_wait_asynccnt)
#    define HAVE_WAIT_ASYNC 1
#  endif
#endif

#if defined(HAVE_WAIT_ASYNC)
#  define WAIT_ASYNC(n) __builtin_amdgcn_s_wait_asynccnt(n)
#elif defined(HAVE_ASYNC_LDS)
#  define WAIT_ASYNC(n) asm volatile("s_wait_asynccnt " #n ::: "memory")
#else
#  define WAIT_ASYNC(n) do {} while (0)
#endif

#if defined(HAVE_ASYNC_LDS)
typedef int v4i __attribute__((ext_vector_type(4)));
typedef __attribute__((address_space(1))) v4i* as1_v4i_p;   // global int4*
typedef __attribute__((address_space(3))) v4i* as3_v4i_p;   // LDS int4*
#endif

// Wave32-parallel digit selection over a 256-bin histogram (run by tid<32).
// Finds largest digit d with count(>d) < rem <= count(>=d); updates shared state.
__device__ __forceinline__ void wave_select_digit(const uint* hist, uint* s_prefix,
                                                  uint* s_remaining, uint* s_eqcnt,
                                                  int shift)
{
    const int lane = (int)(threadIdx.x & 31);
    const uint rem = *s_remaining;

    uint h[8];
#pragma unroll
    for (int j = 0; j < 8; ++j) h[j] = hist[lane * 8 + j];
    uint lanesum = 0;
#pragma unroll
    for (int j = 0; j < 8; ++j) lanesum += h[j];

    // inclusive suffix-sum of lanesum across the wave (5 shuffle steps, no barrier)
    uint v = lanesum;
#pragma unroll
    for (int off = 1; off < 32; off <<= 1) {
        uint t = __shfl_down(v, off, 32);
        if (lane + off < 32) v += t;
    }
    uint c_gt = v - lanesum;   // # elements with digit >= (lane+1)*8, i.e. > my top bin

#pragma unroll
    for (int j = 7; j >= 0; --j) {
        uint c2 = c_gt + h[j];
        if (c_gt < rem && rem <= c2) {     // exactly one (lane, j) block-wide
            *s_remaining = rem - c_gt;
            *s_eqcnt     = h[j];
            *s_prefix   |= ((uint)(lane * 8 + j)) << shift;
        }
        c_gt = c2;
    }
}

__global__ __launch_bounds__(BLOCK) void
topk_scatter_kernel(const float* __restrict__ x, const int* __restrict__ kptr,
                    float* __restrict__ out, int rows)
{
    extern __shared__ unsigned char smem_raw[];
    uint* bufs  = (uint*)smem_raw;            // 2 * COLS   (double-buffered row stage)
    uint* histw = bufs + 2 * COLS;            // NWAVES*256 (per-wave hist, pass 1)
    uint* hist  = histw + NWAVES * 256;       // 256        (shared hist)
    uint* scan  = hist + 256;                 // BLOCK      (tie-rank prefix scan, rare)
    __shared__ uint s_prefix, s_remaining, s_eqcnt;

    const int tid = threadIdx.x;
    uint kk = (uint)kptr[0];
    if (kk > (uint)COLS) kk = (uint)COLS;

    auto issue_row = [&](int row, int buf) {
#if defined(HAVE_ASYNC_LDS)
        const float* g = x + (size_t)row * COLS + tid * EPT;
        uint* l = bufs + buf * COLS + tid * EPT;
#pragma unroll
        for (int i = 0; i < EPT / 4; ++i) {
            __builtin_amdgcn_global_load_async_to_lds_b128(
                (as1_v4i_p)(void*)(g + i * 4),
                (as3_v4i_p)(void*)(l + i * 4),
                0, 0);
        }
#else
        (void)row; (void)buf;
#endif
    };

    int bsel = 0;
#if defined(HAVE_ASYNC_LDS)
    if ((int)blockIdx.x < rows) issue_row((int)blockIdx.x, 0);
#endif

    for (int row = (int)blockIdx.x; row < rows; row += (int)gridDim.x) {
        uint a[EPT];  // this thread's chunk as abs-bit patterns (VGPR resident)

#if defined(HAVE_ASYNC_LDS)
        int nrow = row + (int)gridDim.x;
        if (nrow < rows) { issue_row(nrow, bsel ^ 1); WAIT_ASYNC(4); }
        else             { WAIT_ASYNC(0); }
        {
            uint* l = bufs + bsel * COLS + tid * EPT;
#pragma unroll
            for (int i = 0; i < EPT; ++i) a[i] = l[i] & 0x7fffffffu;
        }
#else
        {
            const uint4* g = (const uint4*)(x + (size_t)row * COLS + tid * EPT);
#pragma unroll
            for (int i = 0; i < EPT / 4; ++i) {
                uint4 v = g[i];
                a[4*i+0] = v.x & 0x7fffffffu;
                a[4*i+1] = v.y & 0x7fffffffu;
                a[4*i+2] = v.z & 0x7fffffffu;
                a[4*i+3] = v.w & 0x7fffffffu;
            }
        }
#endif

        if (kk == 0) {   // degenerate: all zeros
            float4 z = make_float4(0.f, 0.f, 0.f, 0.f);
            float4* orow = (float4*)(out + (size_t)row * COLS + tid * EPT);
#pragma unroll
            for (int i = 0; i < EPT / 4; ++i) orow[i] = z;
            __syncthreads();
            bsel ^= 1;
            continue;
        }

        // ---- radix-select pass 1: digit = bits[31:24], per-wave privatized hist ----
        for (int i = tid; i < NWAVES * 256; i += BLOCK) histw[i] = 0;
        if (tid == 0) { s_prefix = 0u; s_remaining = kk; s_eqcnt = 0u; }
        __syncthreads();
        {
            uint* myh = histw + (tid >> 5) * 256;   // wave32
#pragma unroll
            for (int i = 0; i < EPT; ++i) atomicAdd(&myh[a[i] >> 24], 1u);
        }
        __syncthreads();
        if (tid < 256) {
            uint s = 0;
#pragma unroll
            for (int w = 0; w < NWAVES; ++w) s += histw[w * 256 + tid];
            hist[tid] = s;
        }
        __syncthreads();
        if (tid < 32) wave_select_digit(hist, &s_prefix, &s_remaining, &s_eqcnt, 24);
        __syncthreads();

        // ---- passes 2..4: shifts 16, 8, 0 (few participants; shared hist) ----
#pragma unroll
        for (int shift = 16; shift >= 0; shift -= 8) {
            uint pref = s_prefix;               // stable: prior pass ended with barrier
            if (tid < 256) hist[tid] = 0;
            __syncthreads();
            const uint hi = (uint)(shift + 8);
#pragma unroll
            for (int i = 0; i < EPT; ++i) {
                if ((a[i] >> hi) == (pref >> hi))
                    atomicAdd(&hist[(a[i] >> shift) & 0xFFu], 1u);
            }
            __syncthreads();
            if (tid < 32) wave_select_digit(hist, &s_prefix, &s_remaining, &s_eqcnt, shift);
            __syncthreads();
        }

        const uint T = s_prefix;       // exact abs-bit threshold (k-th largest)
        uint rem_eq  = s_remaining;    // how many ==T to keep
        const uint eqcnt = s_eqcnt;    // how many ==T exist in the row

        uint cnt = 0;
#pragma unroll
        for (int i = 0; i < EPT; ++i) cnt += (a[i] == T) ? 1u : 0u;

        uint excl = 0;
        if (eqcnt != rem_eq) {
            // ---- rare: bit-exact ties straddle the k boundary; stable rank by index ----
            scan[tid] = cnt;
            __syncthreads();
            for (int off = 1; off < BLOCK; off <<= 1) {
                uint v = scan[tid];
                uint add = (tid >= off) ? scan[tid - off] : 0u;
                __syncthreads();
                scan[tid] = v + add;
                __syncthreads();
            }
            excl = scan[tid] - cnt;
        } else {
            rem_eq = 0xFFFFFFFFu;      // keep every ==T element
        }

        // ---- emit: abs value if in top-k else 0 (b128 stores) ----
        float4* orow = (float4*)(out + (size_t)row * COLS + tid * EPT);
#pragma unroll
        for (int i = 0; i < EPT / 4; ++i) {
            float4 o;
            float* op = (float*)&o;
#pragma unroll
            for (int j = 0; j < 4; ++j) {
                uint av = a[4 * i + j];
                float v = 0.0f;
                if (av > T) v = __uint_as_float(av);
                else if (av == T) { if (excl < rem_eq) v = __uint_as_float(av); ++excl; }
                op[j] = v;
            }
            orow[i] = o;
        }

        __syncthreads();   // protect hist/scan/s_* reuse next row
        bsel ^= 1;
    }
}

extern "C" void kernel_launch(void* const* d_in, const int* in_sizes, int n_in,
                              void* d_out, int out_size, void* d_ws, size_t ws_size,
                              hipStream_t stream) {
    (void)n_in; (void)out_size; (void)d_ws; (void)ws_size;
    const float* x  = (const float*)d_in[0];
    const int*   kp = (const int*)d_in[1];
    float*       o  = (float*)d_out;

    const int rows = in_sizes[0] / COLS;
    int grid = rows < 1024 ? rows : 1024;   // persistent blocks -> async double-buffer has work
    if (grid < 1) grid = 1;

    const size_t lds = (size_t)(2 * COLS + NWAVES * 256 + 256 + BLOCK) * sizeof(uint);
    (void)hipFuncSetAttribute((const void*)topk_scatter_kernel,
                              hipFuncAttributeMaxDynamicSharedMemorySize, (int)lds);
    topk_scatter_kernel<<<grid, BLOCK, lds, stream>>>(x, kp, o, rows);
}